// GraphSAGE_26053271617576
// MI455X (gfx1250) — compile-verified
//
#include <hip/hip_runtime.h>
#include <hip/hip_bf16.h>
#include <cstdint>

#define BN_EPS 1e-5f

typedef float v2f __attribute__((ext_vector_type(2)));
typedef float v8f __attribute__((ext_vector_type(8)));

__device__ __forceinline__ v8f wmma4(v2f a, v2f b, v8f c) {
  // D = A(16x4 f32) * B(4x16 f32) + C(16x16 f32), full fp32 precision
  return __builtin_amdgcn_wmma_f32_16x16x4_f32(
      /*neg_a=*/false, a, /*neg_b=*/false, b,
      /*c_mod=*/(short)0, c, /*reuse_a=*/false, /*reuse_b=*/false);
}

// ---------------- utility kernels ----------------

__global__ void fill_f4(float4* __restrict__ p, float v, long long n4) {
  long long i = (long long)blockIdx.x * blockDim.x + threadIdx.x;
  if (i < n4) p[i] = make_float4(v, v, v, v);
}

// Wt[c*K + k] = W[k*C + c]  (column-major weights -> contiguous WMMA B fragments)
__global__ void transpose_w(const float* __restrict__ W, float* __restrict__ Wt,
                            int K, int C) {
  int i = blockIdx.x * blockDim.x + threadIdx.x;
  if (i >= K * C) return;
  int k = i / C;
  int c = i - k * C;
  Wt[(long long)c * K + k] = W[i];
}

__global__ void degree_k(const long long* __restrict__ ei, float* __restrict__ deg, int E) {
  int e = blockIdx.x * blockDim.x + threadIdx.x;
  if (e < E) {
    int dst = (int)ei[(long long)E + e];
    atomicAdd(&deg[dst], 1.0f);
  }
}

__global__ void invdeg_k(float* __restrict__ deg, int n) {
  int i = blockIdx.x * blockDim.x + threadIdx.x;
  if (i < n) deg[i] = 1.0f / fmaxf(deg[i], 1.0f);
}

// agg[dst] += h[src] over all edges; C = 4 << c4log channels, float4 per lane-slot
__global__ __launch_bounds__(256) void scatter_k(const float* __restrict__ h,
                                                 const long long* __restrict__ ei,
                                                 float* __restrict__ agg,
                                                 int E, int c4log) {
  long long tid = (long long)blockIdx.x * blockDim.x + threadIdx.x;
  long long e = tid >> c4log;
  if (e >= E) return;
  int q = (int)(tid & ((1 << c4log) - 1));
  int C = 4 << c4log;
  int src = (int)ei[e];
  int dst = (int)ei[(long long)E + e];
  const float4 v = *(const float4*)(h + (long long)src * C + q * 4);
  float* p = agg + (long long)dst * C + q * 4;
  atomicAdd(p + 0, v.x);
  atomicAdd(p + 1, v.y);
  atomicAdd(p + 2, v.z);
  atomicAdd(p + 3, v.w);
}

// ---------------- fused WMMA GEMM ----------------
// out[M,Cout] = (rs? diag(rs):I) * A1[M,KK] @ W1 (+ A2[M,KK] @ W2) + bias
// W passed TRANSPOSED (Wt[c*KK+k]) so B fragments are contiguous v2f loads at
// compile-time immediate offsets (KK is a template constant).
// Optional BN (g,b,m,v), ReLU, per-column sum (SE squeeze) via atomics.
// One wave computes a 16 x (16*NT) output slab: NT accumulators share each A fragment.
template <int KK, int NT, bool HAS_A2, bool HAS_RS, bool HAS_BN, bool DO_RELU,
          bool DO_COLSUM>
__global__ __launch_bounds__(256) void gemm_wmma(
    const float* __restrict__ A1, const float* __restrict__ rs,
    const float* __restrict__ Wt1, const float* __restrict__ A2,
    const float* __restrict__ Wt2, const float* __restrict__ bias,
    const float* __restrict__ bng, const float* __restrict__ bnb,
    const float* __restrict__ bnm, const float* __restrict__ bnv,
    float* __restrict__ out, float* __restrict__ colsum,
    int M, int Cout) {
  int wave = threadIdx.x >> 5;
  int lane = threadIdx.x & 31;
  int gid = blockIdx.x * 8 + wave;
  int colGroups = Cout / (16 * NT);
  int totalWaves = (M >> 4) * colGroups;
  if (gid >= totalWaves) return;             // wave-uniform exit (EXEC all-1 for WMMA)
  int trow = gid / colGroups;
  int tcol = gid - trow * colGroups;
  int half = lane >> 4;                      // K-split half (f32 WMMA A/B layout)
  int lr = lane & 15;
  int row = trow * 16 + lr;
  int col = tcol * (16 * NT) + lr;

  float rsv = 1.0f;
  if (HAS_RS) rsv = rs[row];

  const float* a1p = A1 + (long long)row * KK + 2 * half;
  const float* a2p = HAS_A2 ? (A2 + (long long)row * KK + 2 * half) : nullptr;
  const float* w1p = Wt1 + (long long)col * KK + 2 * half;
  const float* w2p = HAS_A2 ? (Wt2 + (long long)col * KK + 2 * half) : nullptr;

  v8f acc[NT];
#pragma unroll
  for (int t = 0; t < NT; ++t) acc[t] = (v8f){0.f, 0.f, 0.f, 0.f, 0.f, 0.f, 0.f, 0.f};

#pragma unroll 4
  for (int k = 0; k < KK; k += 4) {
    // ---- issue ALL fragment loads first (clause-able, immediate offsets) ----
    v2f a1 = *(const v2f*)(a1p + k);
    v2f b[NT];
#pragma unroll
    for (int t = 0; t < NT; ++t) b[t] = *(const v2f*)(w1p + 16 * t * KK + k);
    v2f a2 = {0.f, 0.f};
    v2f c2[NT];
    if (HAS_A2) {
      a2 = *(const v2f*)(a2p + k);
#pragma unroll
      for (int t = 0; t < NT; ++t) c2[t] = *(const v2f*)(w2p + 16 * t * KK + k);
    }
    if (HAS_RS) { a1.x *= rsv; a1.y *= rsv; }   // fold inv_deg into A (mean aggr)
    // ---- then the WMMA chain ----
#pragma unroll
    for (int t = 0; t < NT; ++t) acc[t] = wmma4(a1, b[t], acc[t]);
    if (HAS_A2) {
#pragma unroll
      for (int t = 0; t < NT; ++t) acc[t] = wmma4(a2, c2[t], acc[t]);
    }
  }

  // epilogue: +bias, BN, ReLU, store, SE column partial sums
#pragma unroll
  for (int t = 0; t < NT; ++t) {
    int c = col + t * 16;
    float bs = bias[c];
    float sc = 1.0f, sh = 0.0f;
    if (HAS_BN) {
      sc = bng[c] * rsqrtf(bnv[c] + BN_EPS);
      sh = bnb[c] - bnm[c] * sc;
    }
    float partial = 0.0f;
#pragma unroll
    for (int r = 0; r < 8; ++r) {
      float v = acc[t][r] + bs;
      if (HAS_BN) v = v * sc + sh;
      if (DO_RELU) v = fmaxf(v, 0.0f);
      out[(long long)(trow * 16 + r + 8 * half) * Cout + c] = v;
      partial += v;
    }
    if (DO_COLSUM) {
      // lanes l and l+16 hold the same column (rows 0-7 / 8-15): combine, one atomic
      partial += __shfl_xor(partial, 16, 32);
      if (half == 0) atomicAdd(&colsum[c], partial);
    }
  }
}

// ---------------- SE excite (tiny MLP, one block) ----------------
__global__ void se_k(const float* __restrict__ colsum, const float* __restrict__ A,
                     const float* __restrict__ B, float* __restrict__ sevec,
                     int C, int Hh, float invN) {
  __shared__ float ym[256];
  __shared__ float hid[64];
  int t = threadIdx.x;
  if (t < C) ym[t] = colsum[t] * invN;
  __syncthreads();
  if (t < Hh) {
    float s = 0.f;
    for (int c = 0; c < C; ++c) s += ym[c] * A[c * Hh + t];
    hid[t] = fmaxf(s, 0.f);
  }
  __syncthreads();
  if (t < C) {
    float s = 0.f;
    for (int j = 0; j < Hh; ++j) s += hid[j] * B[j * C + t];
    sevec[t] = 1.0f / (1.0f + __expf(-s));
  }
}

// out = relu(in * s[col]) (+ res)
__global__ __launch_bounds__(256) void scale_relu_k(const float* __restrict__ in,
                                                    const float* __restrict__ s,
                                                    const float* __restrict__ res,
                                                    float* __restrict__ out,
                                                    long long n4, int c4log) {
  long long i = (long long)blockIdx.x * blockDim.x + threadIdx.x;
  if (i >= n4) return;
  int c4 = (int)(i & ((1 << c4log) - 1));
  float4 v = ((const float4*)in)[i];
  float4 sv = ((const float4*)s)[c4];
  v.x = fmaxf(v.x * sv.x, 0.f);
  v.y = fmaxf(v.y * sv.y, 0.f);
  v.z = fmaxf(v.z * sv.z, 0.f);
  v.w = fmaxf(v.w * sv.w, 0.f);
  if (res != nullptr) {
    float4 rv = ((const float4*)res)[i];
    v.x += rv.x; v.y += rv.y; v.z += rv.z; v.w += rv.w;
  }
  ((float4*)out)[i] = v;
}

// ---------------- driver ----------------

extern "C" void kernel_launch(void* const* d_in, const int* in_sizes, int n_in,
                              void* d_out, int out_size, void* d_ws, size_t ws_size,
                              hipStream_t stream) {
  const float* x = (const float*)d_in[0];
  const long long* ei = (const long long*)d_in[1];   // int64 edge_index (2,E)
  int p = 2;
  const float* W_in = (const float*)d_in[p++];
  const float* b_in = (const float*)d_in[p++];
  const float* bn0g = (const float*)d_in[p++]; const float* bn0b = (const float*)d_in[p++];
  const float* bn0m = (const float*)d_in[p++]; const float* bn0v = (const float*)d_in[p++];
  const float* Wl0 = (const float*)d_in[p++];
  const float* bl0 = (const float*)d_in[p++];
  const float* Wr0 = (const float*)d_in[p++];
  const float* bn1g = (const float*)d_in[p++]; const float* bn1b = (const float*)d_in[p++];
  const float* bn1m = (const float*)d_in[p++]; const float* bn1v = (const float*)d_in[p++];
  const float* A0 = (const float*)d_in[p++];
  const float* B0 = (const float*)d_in[p++];
  const float* Wl1 = (const float*)d_in[p++];
  const float* bl1 = (const float*)d_in[p++];
  const float* Wr1 = (const float*)d_in[p++];
  const float* bn2g = (const float*)d_in[p++]; const float* bn2b = (const float*)d_in[p++];
  const float* bn2m = (const float*)d_in[p++]; const float* bn2v = (const float*)d_in[p++];
  const float* A1 = (const float*)d_in[p++];
  const float* B1 = (const float*)d_in[p++];
  const float* Wl2 = (const float*)d_in[p++];
  const float* bl2 = (const float*)d_in[p++];
  const float* Wr2 = (const float*)d_in[p++];
  const float* bn3g = (const float*)d_in[p++]; const float* bn3b = (const float*)d_in[p++];
  const float* bn3m = (const float*)d_in[p++]; const float* bn3v = (const float*)d_in[p++];
  const float* W_out = (const float*)d_in[p++];
  const float* b_out = (const float*)d_in[p++];

  const int N = in_sizes[0] / 128;   // 100000
  const int E = in_sizes[1] / 2;     // 1600000
  const float invN = 1.0f / (float)N;

  // workspace layout (floats)
  float* ws = (float*)d_ws;
  long long off = 0;
  float* inv_deg = ws;                  off += ((long long)N + 255) & ~255LL;
  float* colsum = ws + off;             off += 256;
  float* sevec  = ws + off;             off += 256;
  // transposed weights
  float* Wt_in  = ws + off;             off += 128 * 128;
  float* Wtl0   = ws + off;             off += 128 * 256;
  float* Wtr0   = ws + off;             off += 128 * 256;
  float* Wtl1   = ws + off;             off += 256 * 256;
  float* Wtr1   = ws + off;             off += 256 * 256;
  float* Wtl2   = ws + off;             off += 256 * 128;
  float* Wtr2   = ws + off;             off += 256 * 128;
  float* Wt_out = ws + off;             off += 128 * 64;
  off = (off + 255) & ~255LL;
  long long stride = (long long)N * 256;
  float* bufA = ws + off;               // N*256
  float* bufB = bufA + stride;          // N*256
  float* bufC = bufB + stride;          // N*256
  (void)ws_size; (void)n_in; (void)out_size;

  auto blk = [](long long n, int b) { return (unsigned)((n + b - 1) / b); };

  // 0. transpose weights to column-major (contiguous WMMA B fragments)
  transpose_w<<<blk(128 * 128, 256), 256, 0, stream>>>(W_in, Wt_in, 128, 128);
  transpose_w<<<blk(128 * 256, 256), 256, 0, stream>>>(Wl0, Wtl0, 128, 256);
  transpose_w<<<blk(128 * 256, 256), 256, 0, stream>>>(Wr0, Wtr0, 128, 256);
  transpose_w<<<blk(256 * 256, 256), 256, 0, stream>>>(Wl1, Wtl1, 256, 256);
  transpose_w<<<blk(256 * 256, 256), 256, 0, stream>>>(Wr1, Wtr1, 256, 256);
  transpose_w<<<blk(256 * 128, 256), 256, 0, stream>>>(Wl2, Wtl2, 256, 128);
  transpose_w<<<blk(256 * 128, 256), 256, 0, stream>>>(Wr2, Wtr2, 256, 128);
  transpose_w<<<blk(128 * 64, 256), 256, 0, stream>>>(W_out, Wt_out, 128, 64);

  // 1. degree -> inv_deg
  fill_f4<<<blk((long long)N / 4, 256), 256, 0, stream>>>((float4*)inv_deg, 0.f, N / 4);
  degree_k<<<blk(E, 256), 256, 0, stream>>>(ei, inv_deg, E);
  invdeg_k<<<blk(N, 256), 256, 0, stream>>>(inv_deg, N);

  // 2. input_proj: h0 = relu(bn0(x @ W_in + b_in)) -> bufA [N,128]
  {
    int waves = (N >> 4) * (128 / 64);
    gemm_wmma<128, 4, false, false, true, true, false><<<blk(waves, 8), 256, 0, stream>>>(
        x, nullptr, Wt_in, nullptr, nullptr, b_in,
        bn0g, bn0b, bn0m, bn0v, bufA, nullptr, N, 128);
  }

  // 3. layer0: agg = scatter(h0); h = bn1(inv_deg*agg@Wl0 + h0@Wr0 + bl0) -> bufC
  fill_f4<<<blk(((long long)N * 128) / 4, 256), 256, 0, stream>>>(
      (float4*)bufB, 0.f, ((long long)N * 128) / 4);
  scatter_k<<<blk((long long)E << 5, 256), 256, 0, stream>>>(bufA, ei, bufB, E, 5);
  fill_f4<<<1, 64, 0, stream>>>((float4*)colsum, 0.f, 64);
  {
    int waves = (N >> 4) * (256 / 64);
    gemm_wmma<128, 4, true, true, true, false, true><<<blk(waves, 8), 256, 0, stream>>>(
        bufB, inv_deg, Wtl0, bufA, Wtr0, bl0,
        bn1g, bn1b, bn1m, bn1v, bufC, colsum, N, 256);
  }
  se_k<<<1, 256, 0, stream>>>(colsum, A0, B0, sevec, 256, 64, invN);
  // res1 = relu(bufC * se) -> bufA
  scale_relu_k<<<blk(stride / 4, 256), 256, 0, stream>>>(bufC, sevec, nullptr, bufA,
                                                         stride / 4, 6);

  // 4. layer1: agg, gemm, bn2, SE, relu, +res1
  fill_f4<<<blk(stride / 4, 256), 256, 0, stream>>>((float4*)bufB, 0.f, stride / 4);
  scatter_k<<<blk((long long)E << 6, 256), 256, 0, stream>>>(bufA, ei, bufB, E, 6);
  fill_f4<<<1, 64, 0, stream>>>((float4*)colsum, 0.f, 64);
  {
    int waves = (N >> 4) * (256 / 64);
    gemm_wmma<256, 4, true, true, true, false, true><<<blk(waves, 8), 256, 0, stream>>>(
        bufB, inv_deg, Wtl1, bufA, Wtr1, bl1,
        bn2g, bn2b, bn2m, bn2v, bufC, colsum, N, 256);
  }
  se_k<<<1, 256, 0, stream>>>(colsum, A1, B1, sevec, 256, 64, invN);
  // bufC = relu(bufC * se) + bufA(res1)
  scale_relu_k<<<blk(stride / 4, 256), 256, 0, stream>>>(bufC, sevec, bufA, bufC,
                                                         stride / 4, 6);

  // 5. layer2: agg, gemm -> bn3+relu -> bufA [N,128]
  fill_f4<<<blk(stride / 4, 256), 256, 0, stream>>>((float4*)bufB, 0.f, stride / 4);
  scatter_k<<<blk((long long)E << 6, 256), 256, 0, stream>>>(bufC, ei, bufB, E, 6);
  {
    int waves = (N >> 4) * (128 / 64);
    gemm_wmma<256, 4, true, true, true, true, false><<<blk(waves, 8), 256, 0, stream>>>(
        bufB, inv_deg, Wtl2, bufC, Wtr2, bl2,
        bn3g, bn3b, bn3m, bn3v, bufA, nullptr, N, 128);
  }

  // 6. final_proj: out = bufA @ W_out + b_out -> d_out [N,64]
  {
    int waves = (N >> 4) * (64 / 64);
    gemm_wmma<128, 4, false, false, false, false, false><<<blk(waves, 8), 256, 0, stream>>>(
        bufA, nullptr, Wt_out, nullptr, nullptr, b_out,
        nullptr, nullptr, nullptr, nullptr, (float*)d_out, nullptr, N, 64);
  }
}